// StateNetwork_73821897883807
// MI455X (gfx1250) — compile-verified
//
#include <hip/hip_runtime.h>
#include <hip/hip_bf16.h>

#define N_ENT 4096
#define EMB   64
#define NHID  3
#define BATCH 4
#define ALPHA 0.2f
#define MASKF 9e-15f

typedef float v2f __attribute__((ext_vector_type(2)));
typedef float v8f __attribute__((ext_vector_type(8)));
typedef int   v4i __attribute__((ext_vector_type(4)));

// ---------------------------------------------------------------------------
// Kernel 1: h = emb@W (packed {h0,h1,h2,dst}), src, maxdst reduction, row shift M
// Single workgroup, 1024 threads, 4 rows each.
// ---------------------------------------------------------------------------
__global__ __launch_bounds__(1024)
void prep_kernel(const float* __restrict__ emb, const float* __restrict__ W,
                 const float* __restrict__ a,
                 float4* __restrict__ hD, float* __restrict__ srcD,
                 float* __restrict__ MD) {
    __shared__ float Wl[EMB * NHID];
    __shared__ float al[2 * NHID];
    __shared__ float red[32];

    int tid = threadIdx.x;
    if (tid < EMB * NHID) Wl[tid] = W[tid];
    if (tid < 2 * NHID)   al[tid] = a[tid];
    __syncthreads();

    float sva[4];
    float localMax = -3.0e38f;

    #pragma unroll
    for (int t = 0; t < 4; ++t) {
        int row = tid + t * 1024;
        const float4* e4 = (const float4*)(emb + (size_t)row * EMB);
        float h0 = 0.f, h1 = 0.f, h2 = 0.f;
        #pragma unroll
        for (int kk = 0; kk < EMB / 4; ++kk) {
            float4 ev = e4[kk];
            int k = kk * 4;
            h0 = fmaf(ev.x, Wl[(k+0)*3+0], h0); h1 = fmaf(ev.x, Wl[(k+0)*3+1], h1); h2 = fmaf(ev.x, Wl[(k+0)*3+2], h2);
            h0 = fmaf(ev.y, Wl[(k+1)*3+0], h0); h1 = fmaf(ev.y, Wl[(k+1)*3+1], h1); h2 = fmaf(ev.y, Wl[(k+1)*3+2], h2);
            h0 = fmaf(ev.z, Wl[(k+2)*3+0], h0); h1 = fmaf(ev.z, Wl[(k+2)*3+1], h1); h2 = fmaf(ev.z, Wl[(k+2)*3+2], h2);
            h0 = fmaf(ev.w, Wl[(k+3)*3+0], h0); h1 = fmaf(ev.w, Wl[(k+3)*3+1], h1); h2 = fmaf(ev.w, Wl[(k+3)*3+2], h2);
        }
        float sv = h0 * al[0] + h1 * al[1] + h2 * al[2];
        float dv = h0 * al[3] + h1 * al[4] + h2 * al[5];
        hD[row]   = make_float4(h0, h1, h2, dv);
        srcD[row] = sv;
        sva[t]    = sv;
        localMax  = fmaxf(localMax, dv);
    }

    // block-wide max reduction (wave32)
    int lane = tid & 31, wave = tid >> 5;
    #pragma unroll
    for (int off = 16; off > 0; off >>= 1)
        localMax = fmaxf(localMax, __shfl_xor(localMax, off, 32));
    if (lane == 0) red[wave] = localMax;
    __syncthreads();
    if (tid < 32) {
        float m = red[tid];
        #pragma unroll
        for (int off = 16; off > 0; off >>= 1)
            m = fmaxf(m, __shfl_xor(m, off, 32));
        if (tid == 0) red[0] = m;
    }
    __syncthreads();
    float maxdst = red[0];

    #pragma unroll
    for (int t = 0; t < 4; ++t) {
        int row = tid + t * 1024;
        float v = sva[t] + maxdst;
        v = (v >= 0.f) ? v : ALPHA * v;      // leakyrelu upper bound of row max
        MD[row] = fmaxf(v, MASKF);
    }
}

// ---------------------------------------------------------------------------
// Kernel 2: fused masked-softmax + att@h + ELU.  One wave per row (8 rows/wave),
// adj streamed with coalesced NON-TEMPORAL int4 loads (read-once 256MB; keep it
// out of L2); {h0,h1,h2,dst} table staged once in LDS (one b128 DS read per j).
// grid = 256 blocks x 256 threads (8 waves). 64 rows per block.
// ---------------------------------------------------------------------------
__global__ __launch_bounds__(256)
void gat_rows_kernel(const int* __restrict__ adj, const float4* __restrict__ hD,
                     const float* __restrict__ srcD, const float* __restrict__ MD,
                     float* __restrict__ x) {
    __shared__ float4 hs[N_ENT];   // 64 KB
    int tid = threadIdx.x;
    for (int t = tid; t < N_ENT; t += 256) hs[t] = hD[t];
    __syncthreads();

    int lane = tid & 31, wave = tid >> 5;
    int rowBase = blockIdx.x * 64 + wave * 8;   // global row index (b*4096 + i)

    for (int r = 0; r < 8; ++r) {
        int grow = rowBase + r;           // 0 .. 16383
        int i    = grow & (N_ENT - 1);
        int b    = grow >> 12;
        const int* arow = adj + (size_t)grow * N_ENT;   // adj[b, i, :]
        float si = srcD[i];
        float Mi = MD[i];

        float s = 0.f, a0 = 0.f, a1 = 0.f, a2 = 0.f;
        for (int j0 = lane * 4; j0 < N_ENT; j0 += 128) {
            __builtin_prefetch(arow + j0 + 512, 0, 1);   // -> global_prefetch_b8
            v4i av = __builtin_nontemporal_load((const v4i*)(arow + j0));
            #pragma unroll
            for (int t = 0; t < 4; ++t) {
                float4 hv = hs[j0 + t];
                float v = si + hv.w;
                v = (v >= 0.f) ? v : ALPHA * v;          // leakyrelu
                v = (av[t] > 0) ? v : MASKF;             // 9e-15 fill (still in softmax!)
                float w = __expf(v - Mi);
                s  += w;
                a0 = fmaf(w, hv.x, a0);
                a1 = fmaf(w, hv.y, a1);
                a2 = fmaf(w, hv.z, a2);
            }
        }
        // wave32 reduction
        #pragma unroll
        for (int off = 16; off > 0; off >>= 1) {
            s  += __shfl_xor(s,  off, 32);
            a0 += __shfl_xor(a0, off, 32);
            a1 += __shfl_xor(a1, off, 32);
            a2 += __shfl_xor(a2, off, 32);
        }
        if (lane == 0) {
            float inv = 1.f / s;
            float h0 = a0 * inv, h1 = a1 * inv, h2 = a2 * inv;
            h0 = (h0 > 0.f) ? h0 : (__expf(h0) - 1.f);   // ELU (alpha=1)
            h1 = (h1 > 0.f) ? h1 : (__expf(h1) - 1.f);
            h2 = (h2 > 0.f) ? h2 : (__expf(h2) - 1.f);
            float* xp = x + (size_t)b * (N_ENT * NHID) + (size_t)i * NHID;
            xp[0] = h0; xp[1] = h1; xp[2] = h2;
        }
    }
}

// ---------------------------------------------------------------------------
// Kernel 3: partial = x[4,12288] @ fc1_w^T   via V_WMMA_F32_16X16X4_F32.
// Branchless: pad rows load from clamped addresses, zeroed via 0/1 mask mul,
// each fragment half is one global_load_b64.  EXEC stays all-1s in the loop.
// A layout (16x4 f32): lane = m + 16*(k/2), vgpr = k%2.
// B layout (4x16 f32): lane = n + 16*(k/2), vgpr = k%2.
// D layout (16x16 f32): vgpr v: lanes0-15 M=v, lanes16-31 M=v+8; N=lane%16.
// ---------------------------------------------------------------------------
__global__ __launch_bounds__(256)
void fc_wmma_kernel(const float* __restrict__ x, const float* __restrict__ fw,
                    float* __restrict__ partial) {
    int tid = threadIdx.x, lane = tid & 31, wave = tid >> 5;
    if (wave >= 7) return;                     // whole wave exits; others full-EXEC
    const int K = N_ENT * NHID;                // 12288
    int nb    = wave * 16;
    int kbase = blockIdx.x * 128;
    int m     = lane & 15;
    int khalf = (lane >> 4) * 2;               // 0 or 2
    int n     = nb + m;
    int mc    = m & (BATCH - 1);               // clamped (valid) A row
    int nc    = (n < 100) ? n : 99;            // clamped (valid) B row
    float mmask = (m < BATCH) ? 1.f : 0.f;
    float nmask = (n < 100)   ? 1.f : 0.f;

    const v2f* xp = (const v2f*)(x  + (size_t)mc * K + kbase + khalf);
    const v2f* wp = (const v2f*)(fw + (size_t)nc * K + kbase + khalf);

    v8f c = {0.f, 0.f, 0.f, 0.f, 0.f, 0.f, 0.f, 0.f};
    #pragma unroll 4
    for (int ks = 0; ks < 32; ++ks) {
        v2f av = xp[ks * 2];                   // k advances by 4 floats = 2 v2f
        v2f bv = wp[ks * 2];
        v2f A = { av.x * mmask, av.y * mmask };
        v2f B = { bv.x * nmask, bv.y * nmask };
        c = __builtin_amdgcn_wmma_f32_16x16x4_f32(false, A, false, B,
                                                  (short)0, c, false, false);
    }
    // write partial D: lanes 0-15 hold M=v (0..7); only M<4 valid
    if (n < 100) {
        int Mbase = (lane >> 4) * 8;
        float* p = partial + (size_t)blockIdx.x * (BATCH * 100);
        #pragma unroll
        for (int v = 0; v < 8; ++v) {
            int M = Mbase + v;
            if (M < BATCH) p[M * 100 + n] = c[v];
        }
    }
}

// ---------------------------------------------------------------------------
// Kernel 4: sum 96 partials + bias -> out[4,100]  (deterministic reduce)
// ---------------------------------------------------------------------------
__global__ __launch_bounds__(256)
void finalize_kernel(const float* __restrict__ partial, const float* __restrict__ bias,
                     float* __restrict__ out) {
    int t = blockIdx.x * blockDim.x + threadIdx.x;
    if (t < BATCH * 100) {
        int nn = t % 100;
        float s = bias[nn];
        for (int sl = 0; sl < 96; ++sl) s += partial[(size_t)sl * (BATCH * 100) + t];
        out[t] = s;
    }
}

// ---------------------------------------------------------------------------
extern "C" void kernel_launch(void* const* d_in, const int* in_sizes, int n_in,
                              void* d_out, int out_size, void* d_ws, size_t ws_size,
                              hipStream_t stream) {
    const int*   adj   = (const int*)  d_in[0];   // [4,4096,4096] int32
    const float* emb   = (const float*)d_in[1];   // [4096,64]
    const float* W     = (const float*)d_in[2];   // [64,3]
    const float* a     = (const float*)d_in[3];   // [6,1]
    const float* fc1_w = (const float*)d_in[4];   // [100,12288]
    const float* fc1_b = (const float*)d_in[5];   // [100]
    float*       out   = (float*)d_out;           // [4,100]

    char* ws = (char*)d_ws;
    float4* hD      = (float4*)(ws + 0);          //  65536 B
    float*  srcD    = (float*) (ws + 65536);      //  16384 B
    float*  MD      = (float*) (ws + 81920);      //  16384 B
    float*  x       = (float*) (ws + 98304);      // 196608 B  [4,12288]
    float*  partial = (float*) (ws + 294912);     // 153600 B  [96,400]

    prep_kernel<<<1, 1024, 0, stream>>>(emb, W, a, hD, srcD, MD);
    gat_rows_kernel<<<256, 256, 0, stream>>>(adj, hD, srcD, MD, x);
    fc_wmma_kernel<<<96, 256, 0, stream>>>(x, fc1_w, partial);
    finalize_kernel<<<2, 256, 0, stream>>>(partial, fc1_b, out);
}